// ECC_50861002719987
// MI455X (gfx1250) — compile-verified
//
#include <hip/hip_runtime.h>
#include <hip/hip_bf16.h>
#include <math.h>

#define N_NODES 20000
#define N_EDGES 320000
#define N_TILES (N_NODES / 16)   // 1250
#define E_TILES (N_EDGES / 16)   // 20000

typedef __attribute__((ext_vector_type(2))) float v2f;
typedef __attribute__((ext_vector_type(8))) float v8f;

static __device__ __forceinline__ v8f wmma_f32_4(v2f a, v2f b, v8f c) {
  // D = A(16x4,f32) * B(4x16,f32) + C(16x16,f32)
  return __builtin_amdgcn_wmma_f32_16x16x4_f32(false, a, false, b, (short)0, c,
                                               false, false);
}

// ---------------------------------------------------------------------------
// Zero-fill scratch accumulators
// ---------------------------------------------------------------------------
__global__ void ecc_zero_kernel(float* __restrict__ p, int n) {
  int i = blockIdx.x * blockDim.x + threadIdx.x;
  int stride = gridDim.x * blockDim.x;
  for (; i < n; i += stride) p[i] = 0.0f;
}

// ---------------------------------------------------------------------------
// h = x @ W_pre + b_pre        ([20000,64] @ [64,16])
// One wave handles 16 node rows; K=64 -> 16 WMMA f32 16x16x4 steps.
// ---------------------------------------------------------------------------
__global__ __launch_bounds__(256) void ecc_pre_kernel(
    const float* __restrict__ x, const float* __restrict__ Wp,
    const float* __restrict__ bp, float* __restrict__ h) {
  const int lane = threadIdx.x & 31;
  const int laneLo = lane & 15;
  const int laneHi = lane >> 4;
  const int tile = blockIdx.x * (blockDim.x >> 5) + (threadIdx.x >> 5);
  if (tile >= N_TILES) return;  // wave-uniform guard (EXEC stays all-ones)
  const int row0 = tile * 16;

  v8f acc = {};
#pragma unroll
  for (int kk = 0; kk < 16; ++kk) {
    const int c0 = 4 * kk + 2 * laneHi;
    const float* px = x + (size_t)(row0 + laneLo) * 64 + c0;
    v2f a; a.x = px[0]; a.y = px[1];
    v2f b; b.x = Wp[c0 * 16 + laneLo]; b.y = Wp[c0 * 16 + 16 + laneLo];
    acc = wmma_f32_4(a, b, acc);
  }
  const float bias = bp[laneLo];
#pragma unroll
  for (int r = 0; r < 8; ++r)
    h[(size_t)(row0 + 8 * laneHi + r) * 16 + laneLo] = acc[r] + bias;
}

// ---------------------------------------------------------------------------
// Fused NNConv edge pass:
//   m      = relu(edge_attr @ W1 + b1)                        [16 per edge]
//   msg[o] = sum_{k,i} m[k]*h_src[i]*W2[k,i*16+o] + sum_i h_src[i]*b2[i*16+o]
//   agg[dst] += msg      (atomic scatter-add)
// Wave processes 16 edges: A = outer-product matrix p (16x272, bias-augmented),
// B = reshaped [272,16] weight (in LDS, shared by block) -> 68 WMMA steps.
// ---------------------------------------------------------------------------
__global__ __launch_bounds__(256) void ecc_edge_kernel(
    const float* __restrict__ hin, const int* __restrict__ eidx,
    const float* __restrict__ eattr, const float* __restrict__ W1,
    const float* __restrict__ b1, const float* __restrict__ W2,
    const float* __restrict__ b2, float* __restrict__ agg) {
  __shared__ float sB[16 * 276];  // Bt[o][c], c-stride 276 (bank-conflict pad)
  __shared__ float sW1[48];
  __shared__ float sb1[16];

  const int tid = threadIdx.x;
  for (int i = tid; i < 16 * 272; i += 256) {
    const int o = i / 272, c = i % 272;
    float v;
    if (c < 256) v = W2[(c >> 4) * 256 + ((c & 15) << 4) + o];
    else         v = b2[((c - 256) << 4) + o];
    sB[o * 276 + c] = v;
  }
  if (tid < 48) sW1[tid] = W1[tid];
  if (tid < 16) sb1[tid] = b1[tid];
  __syncthreads();

  const int lane = tid & 31;
  const int laneLo = lane & 15;
  const int laneHi = lane >> 4;
  const int tile = blockIdx.x * 8 + (tid >> 5);
  const int e = tile * 16 + laneLo;  // this lane's edge (both halves mirror)

  const int src = eidx[e];
  const int dl  = eidx[N_EDGES + e];

  // edge MLP layer 1 (3 -> 16) + ReLU
  const float a0 = eattr[(size_t)e * 3 + 0];
  const float a1 = eattr[(size_t)e * 3 + 1];
  const float a2 = eattr[(size_t)e * 3 + 2];
  float mreg[16];
#pragma unroll
  for (int k = 0; k < 16; ++k) {
    float t = sb1[k] + a0 * sW1[k] + a1 * sW1[16 + k] + a2 * sW1[32 + k];
    mreg[k] = t > 0.0f ? t : 0.0f;
  }

  // source node features (16 floats)
  const float4* ph = reinterpret_cast<const float4*>(hin + (size_t)src * 16);
  const float4 h0 = ph[0], h1 = ph[1], h2 = ph[2], h3 = ph[3];
  const float hv[16] = {h0.x, h0.y, h0.z, h0.w, h1.x, h1.y, h1.z, h1.w,
                        h2.x, h2.y, h2.z, h2.w, h3.x, h3.y, h3.z, h3.w};
  // pre-select the laneHi-dependent h pair per K sub-block (compile-time idx)
  float hs0[4], hs1[4];
#pragma unroll
  for (int q = 0; q < 4; ++q) {
    hs0[q] = laneHi ? hv[4 * q + 2] : hv[4 * q + 0];
    hs1[q] = laneHi ? hv[4 * q + 3] : hv[4 * q + 1];
  }

  const float* bRow = sB + laneLo * 276 + 2 * laneHi;
  v8f acc = {};
#pragma unroll
  for (int kk = 0; kk < 64; ++kk) {          // p[c] = m[c>>4] * h[c&15]
    const int k = kk >> 2, q = kk & 3;
    v2f a; a.x = mreg[k] * hs0[q]; a.y = mreg[k] * hs1[q];
    v2f b; b.x = bRow[4 * kk]; b.y = bRow[4 * kk + 1];
    acc = wmma_f32_4(a, b, acc);
  }
#pragma unroll
  for (int q = 0; q < 4; ++q) {              // bias rows: p[256+i] = h[i]
    v2f a; a.x = hs0[q]; a.y = hs1[q];
    v2f b; b.x = bRow[256 + 4 * q]; b.y = bRow[256 + 4 * q + 1];
    acc = wmma_f32_4(a, b, acc);
  }

  // scatter-add: acc[r] = msg[edge 8*laneHi+r][col laneLo]
#pragma unroll
  for (int r = 0; r < 8; ++r) {
    const int de = __shfl(dl, 8 * laneHi + r, 32);
    atomicAdd(&agg[(size_t)de * 16 + laneLo], acc[r]);
  }
}

// ---------------------------------------------------------------------------
// Node update: out = agg + hin @ root + bias, then ReLU (mode 0) or
// row L2-normalize (mode 1). K=16 -> 4 WMMA steps; C seeded from agg.
// ---------------------------------------------------------------------------
__global__ __launch_bounds__(256) void ecc_node_kernel(
    const float* __restrict__ hin, const float* __restrict__ root,
    const float* __restrict__ bias, const float* __restrict__ agg,
    float* __restrict__ out, int mode) {
  const int lane = threadIdx.x & 31;
  const int laneLo = lane & 15;
  const int laneHi = lane >> 4;
  const int tile = blockIdx.x * (blockDim.x >> 5) + (threadIdx.x >> 5);
  if (tile >= N_TILES) return;
  const int row0 = tile * 16;

  v8f acc;
#pragma unroll
  for (int r = 0; r < 8; ++r)
    acc[r] = agg[(size_t)(row0 + 8 * laneHi + r) * 16 + laneLo];

#pragma unroll
  for (int kk = 0; kk < 4; ++kk) {
    const int c0 = 4 * kk + 2 * laneHi;
    const float* px = hin + (size_t)(row0 + laneLo) * 16 + c0;
    v2f a; a.x = px[0]; a.y = px[1];
    v2f b; b.x = root[c0 * 16 + laneLo]; b.y = root[c0 * 16 + 16 + laneLo];
    acc = wmma_f32_4(a, b, acc);
  }

  const float bv = bias[laneLo];
  if (mode == 0) {
#pragma unroll
    for (int r = 0; r < 8; ++r) {
      const float v = acc[r] + bv;
      out[(size_t)(row0 + 8 * laneHi + r) * 16 + laneLo] = v > 0.0f ? v : 0.0f;
    }
  } else {
#pragma unroll
    for (int r = 0; r < 8; ++r) {
      const float v = acc[r] + bv;
      float ss = v * v;                 // 16-lane row reduction (per half)
      ss += __shfl_xor(ss, 1, 32);
      ss += __shfl_xor(ss, 2, 32);
      ss += __shfl_xor(ss, 4, 32);
      ss += __shfl_xor(ss, 8, 32);
      const float n = sqrtf(ss);
      out[(size_t)(row0 + 8 * laneHi + r) * 16 + laneLo] =
          v / fmaxf(n, 1e-12f);
    }
  }
}

// ---------------------------------------------------------------------------
extern "C" void kernel_launch(void* const* d_in, const int* in_sizes, int n_in,
                              void* d_out, int out_size, void* d_ws,
                              size_t ws_size, hipStream_t stream) {
  const float* x     = (const float*)d_in[0];
  const int*   eidx  = (const int*)  d_in[1];
  const float* eattr = (const float*)d_in[2];
  const float* Wpre  = (const float*)d_in[3];
  const float* bpre  = (const float*)d_in[4];
  const float* e1W1  = (const float*)d_in[5];
  const float* e1b1  = (const float*)d_in[6];
  const float* e1W2  = (const float*)d_in[7];
  const float* e1b2  = (const float*)d_in[8];
  const float* root1 = (const float*)d_in[9];
  const float* bias1 = (const float*)d_in[10];
  const float* e2W1  = (const float*)d_in[11];
  const float* e2b1  = (const float*)d_in[12];
  const float* e2W2  = (const float*)d_in[13];
  const float* e2b2  = (const float*)d_in[14];
  const float* root2 = (const float*)d_in[15];
  const float* bias2 = (const float*)d_in[16];

  float* ws   = (float*)d_ws;
  float* h    = ws;                 // [20000,16]
  float* h2   = ws + 320000;        // [20000,16]
  float* agg1 = ws + 640000;        // [20000,16]
  float* agg2 = ws + 960000;        // [20000,16] (contiguous with agg1)
  float* out  = (float*)d_out;

  const int nodeBlocks = (N_TILES + 7) / 8;  // 8 waves/block

  ecc_zero_kernel<<<2500, 256, 0, stream>>>(agg1, 2 * N_NODES * 16);
  ecc_pre_kernel<<<nodeBlocks, 256, 0, stream>>>(x, Wpre, bpre, h);
  ecc_edge_kernel<<<E_TILES / 8, 256, 0, stream>>>(h, eidx, eattr, e1W1, e1b1,
                                                   e1W2, e1b2, agg1);
  ecc_node_kernel<<<nodeBlocks, 256, 0, stream>>>(h, root1, bias1, agg1, h2, 0);
  ecc_edge_kernel<<<E_TILES / 8, 256, 0, stream>>>(h2, eidx, eattr, e2W1, e2b1,
                                                   e2W2, e2b2, agg2);
  ecc_node_kernel<<<nodeBlocks, 256, 0, stream>>>(h2, root2, bias2, agg2, out, 1);
}